// GIN_3layer_48266842472562
// MI455X (gfx1250) — compile-verified
//
#include <hip/hip_runtime.h>
#include <stdint.h>

typedef float v2f __attribute__((ext_vector_type(2)));
typedef float v8f __attribute__((ext_vector_type(8)));

#define LP 132          // LDS row pitch in floats (128 + 4; 528B is 16B-aligned)
#define BN_EPS 1e-5f

// ---------------------------------------------------------------------------
// float4 copy (grid-stride)
__global__ void copy4_kernel(const float4* __restrict__ in, float4* __restrict__ out,
                             long long n4) {
    long long i = blockIdx.x * (long long)blockDim.x + threadIdx.x;
    long long stride = gridDim.x * (long long)blockDim.x;
    for (; i < n4; i += stride) out[i] = in[i];
}

// zero small buffer
__global__ void zero_kernel(float* __restrict__ p, int n) {
    int i = blockIdx.x * blockDim.x + threadIdx.x;
    if (i < n) p[i] = 0.f;
}

// ---------------------------------------------------------------------------
// Edge scatter-add: one wave per edge, lane handles 4 consecutive features.
// agg must be pre-initialized with the self term. L2-resident float atomics.
__global__ void edge_agg_kernel(const float* __restrict__ feat,
                                const long long* __restrict__ src,
                                const long long* __restrict__ dst,
                                float* __restrict__ agg, long long E) {
    int lane = threadIdx.x & 31;
    long long wid = ((long long)blockIdx.x * blockDim.x + threadIdx.x) >> 5;
    long long nwaves = ((long long)gridDim.x * blockDim.x) >> 5;
    for (long long e = wid; e < E; e += nwaves) {
        long long s = src[e];
        long long d = dst[e];
        float4 v = ((const float4*)(feat + s * 128))[lane];
        float* p = agg + d * 128 + lane * 4;
        atomicAdd(p + 0, v.x);
        atomicAdd(p + 1, v.y);
        atomicAdd(p + 2, v.z);
        atomicAdd(p + 3, v.w);
    }
}

// ---------------------------------------------------------------------------
// Fused GEMM + bias + BN-stat accumulation.
// H = A[N,128] @ W[128,128]^T + bias ; stats[0..127]+=col sums, [128..255]+=col sumsq
// Block: 256 threads = 8 waves; block tile = 128 rows x 128 cols; K = 128.
// Uses V_WMMA_F32_16X16X4_F32 (fp32, matches reference numerics).
__global__ __launch_bounds__(256)
void gemm_bn_kernel(const float* __restrict__ A, const float* __restrict__ W,
                    const float* __restrict__ bias, float* __restrict__ Hout,
                    float* __restrict__ stats, int N) {
    extern __shared__ float smem[];
    float* Wl = smem;              // [128][LP]  W in native [n][k] layout
    float* Al = smem + 128 * LP;   // [128][LP]

    const int tid = threadIdx.x;
    const int rowBase = blockIdx.x * 128;

    // Stage W (128x128 floats) coalesced, float4.
    for (int i = tid; i < 128 * 32; i += 256) {
        int n = i >> 5, kq = i & 31;
        float4 w4 = ((const float4*)W)[i];
        *(float4*)&Wl[n * LP + kq * 4] = w4;
    }
    // Stage A tile (128 rows), zero-pad OOB rows so WMMA runs full-EXEC.
    for (int i = tid; i < 128 * 32; i += 256) {
        int r = i >> 5, kq = i & 31;
        int row = rowBase + r;
        float4 a4 = make_float4(0.f, 0.f, 0.f, 0.f);
        if (row < N) a4 = ((const float4*)A)[(long long)row * 32 + kq];
        *(float4*)&Al[r * LP + kq * 4] = a4;
    }
    __syncthreads();

    const int wave = tid >> 5;
    const int lane = tid & 31;
    const int m16  = lane & 15;
    const int half = lane >> 4;
    const float* Arow = &Al[(wave * 16 + m16) * LP];

    v8f acc[8] = {};
    for (int kt = 0; kt < 32; ++kt) {
        const int k0 = kt * 4 + half * 2;
        v2f a;
        a.x = Arow[k0];
        a.y = Arow[k0 + 1];
#pragma unroll
        for (int nt = 0; nt < 8; ++nt) {
            const int n = nt * 16 + m16;
            v2f b;
            b.x = Wl[n * LP + k0];
            b.y = Wl[n * LP + k0 + 1];
            acc[nt] = __builtin_amdgcn_wmma_f32_16x16x4_f32(
                false, a, false, b, (short)0, acc[nt], false, false);
        }
    }

    // Epilogue: bias, store pre-BN activations, accumulate BN stats.
#pragma unroll
    for (int nt = 0; nt < 8; ++nt) {
        const int n = nt * 16 + m16;
        const float bi = bias[n];
        float s = 0.f, sq = 0.f;
#pragma unroll
        for (int v = 0; v < 8; ++v) {
            int row = rowBase + wave * 16 + half * 8 + v;
            if (row < N) {
                float val = acc[nt][v] + bi;
                Hout[(long long)row * 128 + n] = val;
                s += val;
                sq += val * val;
            }
        }
        atomicAdd(&stats[n], s);
        atomicAdd(&stats[128 + n], sq);
    }
}

// ---------------------------------------------------------------------------
// BN finalize: stats[256+c]=scale, stats[384+c]=shift
__global__ void bn_finalize_kernel(float* __restrict__ stats,
                                   const float* __restrict__ gamma,
                                   const float* __restrict__ beta, float invN) {
    int c = threadIdx.x;
    if (c < 128) {
        float mean = stats[c] * invN;
        float var  = stats[128 + c] * invN - mean * mean;
        float inv  = rsqrtf(var + BN_EPS);
        float sc   = gamma[c] * inv;
        stats[256 + c] = sc;
        stats[384 + c] = beta[c] - mean * sc;
    }
}

// BN + ReLU, dual-write: post (in-place over h is fine) and next-layer agg init.
__global__ void bn_relu_dual_kernel(const float* __restrict__ h,
                                    const float* __restrict__ stats,
                                    float* __restrict__ post,
                                    float* __restrict__ aggNext, long long n4) {
    const float* scale = stats + 256;
    const float* shift = stats + 384;
    long long i = blockIdx.x * (long long)blockDim.x + threadIdx.x;
    long long stride = gridDim.x * (long long)blockDim.x;
    for (; i < n4; i += stride) {
        int c = (int)(i & 31) * 4;
        float4 v = ((const float4*)h)[i];
        float4 r;
        r.x = fmaxf(v.x * scale[c + 0] + shift[c + 0], 0.f);
        r.y = fmaxf(v.y * scale[c + 1] + shift[c + 1], 0.f);
        r.z = fmaxf(v.z * scale[c + 2] + shift[c + 2], 0.f);
        r.w = fmaxf(v.w * scale[c + 3] + shift[c + 3], 0.f);
        ((float4*)post)[i] = r;
        ((float4*)aggNext)[i] = r;
    }
}

// ---------------------------------------------------------------------------
// Final layer: out[N,10] = A[N,128] @ W3[10,128]^T + b3. One 16-wide col tile
// (cols 10..15 zero-padded in LDS); stores guarded to n<10.
__global__ __launch_bounds__(256)
void gemm_out_kernel(const float* __restrict__ A, const float* __restrict__ W3,
                     const float* __restrict__ b3, float* __restrict__ out, int N) {
    extern __shared__ float smem[];
    float* Wl = smem;             // [16][LP]
    float* Al = smem + 16 * LP;   // [128][LP]

    const int tid = threadIdx.x;
    const int rowBase = blockIdx.x * 128;

    for (int i = tid; i < 16 * 32; i += 256) {
        int n = i >> 5, kq = i & 31;
        float4 w4 = make_float4(0.f, 0.f, 0.f, 0.f);
        if (n < 10) w4 = ((const float4*)W3)[n * 32 + kq];
        *(float4*)&Wl[n * LP + kq * 4] = w4;
    }
    for (int i = tid; i < 128 * 32; i += 256) {
        int r = i >> 5, kq = i & 31;
        int row = rowBase + r;
        float4 a4 = make_float4(0.f, 0.f, 0.f, 0.f);
        if (row < N) a4 = ((const float4*)A)[(long long)row * 32 + kq];
        *(float4*)&Al[r * LP + kq * 4] = a4;
    }
    __syncthreads();

    const int wave = tid >> 5;
    const int lane = tid & 31;
    const int m16  = lane & 15;
    const int half = lane >> 4;
    const float* Arow = &Al[(wave * 16 + m16) * LP];

    v8f acc = {};
    for (int kt = 0; kt < 32; ++kt) {
        const int k0 = kt * 4 + half * 2;
        v2f a;
        a.x = Arow[k0];
        a.y = Arow[k0 + 1];
        v2f b;
        b.x = Wl[m16 * LP + k0];
        b.y = Wl[m16 * LP + k0 + 1];
        acc = __builtin_amdgcn_wmma_f32_16x16x4_f32(
            false, a, false, b, (short)0, acc, false, false);
    }

    if (m16 < 10) {
        const float bi = b3[m16];
#pragma unroll
        for (int v = 0; v < 8; ++v) {
            int row = rowBase + wave * 16 + half * 8 + v;
            if (row < N) out[(long long)row * 10 + m16] = acc[v] + bi;
        }
    }
}

// ---------------------------------------------------------------------------
extern "C" void kernel_launch(void* const* d_in, const int* in_sizes, int n_in,
                              void* d_out, int out_size, void* d_ws, size_t ws_size,
                              hipStream_t stream) {
    const float*     x   = (const float*)d_in[0];
    const long long* ei  = (const long long*)d_in[1];   // int64 [2,E]
    const float*     W1  = (const float*)d_in[2];
    const float*     b1  = (const float*)d_in[3];
    const float*     g1  = (const float*)d_in[4];
    const float*     be1 = (const float*)d_in[5];
    const float*     W2  = (const float*)d_in[6];
    const float*     b2  = (const float*)d_in[7];
    const float*     g2  = (const float*)d_in[8];
    const float*     be2 = (const float*)d_in[9];
    const float*     W3  = (const float*)d_in[10];
    const float*     b3  = (const float*)d_in[11];
    float*           out = (float*)d_out;

    const int       N = in_sizes[0] / 128;
    const long long E = (long long)in_sizes[1] / 2;
    const long long NF  = (long long)N * 128;
    const long long NF4 = NF / 4;

    float* buf0  = (float*)d_ws;            // [N,128]
    float* buf1  = buf0 + NF;               // [N,128]
    float* stats = buf1 + NF;               // 512 floats: sum, sumsq, scale, shift

    const long long* src = ei;
    const long long* dst = ei + E;

    const int ELT_BLOCKS  = 4096;
    const int EDGE_BLOCKS = 8192;
    const int GEMM_BLOCKS = (N + 127) / 128;
    const size_t SMEM_GEMM = (size_t)(128 + 128) * LP * sizeof(float);
    const size_t SMEM_OUT  = (size_t)(16 + 128) * LP * sizeof(float);
    const float invN = 1.0f / (float)N;

    // ---- Layer 1 ----
    copy4_kernel<<<ELT_BLOCKS, 256, 0, stream>>>((const float4*)x, (float4*)buf0, NF4);
    edge_agg_kernel<<<EDGE_BLOCKS, 256, 0, stream>>>(x, src, dst, buf0, E);
    zero_kernel<<<1, 256, 0, stream>>>(stats, 256);
    gemm_bn_kernel<<<GEMM_BLOCKS, 256, SMEM_GEMM, stream>>>(buf0, W1, b1, buf0, stats, N);
    bn_finalize_kernel<<<1, 128, 0, stream>>>(stats, g1, be1, invN);
    bn_relu_dual_kernel<<<ELT_BLOCKS, 256, 0, stream>>>(buf0, stats, buf0, buf1, NF4);

    // ---- Layer 2 ----
    edge_agg_kernel<<<EDGE_BLOCKS, 256, 0, stream>>>(buf0, src, dst, buf1, E);
    zero_kernel<<<1, 256, 0, stream>>>(stats, 256);
    gemm_bn_kernel<<<GEMM_BLOCKS, 256, SMEM_GEMM, stream>>>(buf1, W2, b2, buf1, stats, N);
    bn_finalize_kernel<<<1, 128, 0, stream>>>(stats, g2, be2, invN);
    bn_relu_dual_kernel<<<ELT_BLOCKS, 256, 0, stream>>>(buf1, stats, buf1, buf0, NF4);

    // ---- Layer 3 ----
    edge_agg_kernel<<<EDGE_BLOCKS, 256, 0, stream>>>(buf1, src, dst, buf0, E);
    gemm_out_kernel<<<GEMM_BLOCKS, 256, SMEM_OUT, stream>>>(buf0, W3, b3, out, N);

    (void)n_in; (void)out_size; (void)ws_size;
}